// ChamfersDistance_23373212024987
// MI455X (gfx1250) — compile-verified
//
#include <hip/hip_runtime.h>

typedef __attribute__((ext_vector_type(2))) float v2f;
typedef __attribute__((ext_vector_type(8))) float v8f;

#define B_SZ 4
#define N_SZ 8192
#define M_SZ 8192
#define CHUNK 1024                        // Y points per segment (blockIdx.z)
#define YSEG  (M_SZ / CHUNK)              // 8 segments
#define WAVES_PER_WG 8                    // 256 threads, wave32
#define TPW 4                             // X tiles (16 rows) per wave
#define ROWS_PER_WG (WAVES_PER_WG * TPW * 16)   // 512
#define GRID_X (N_SZ / ROWS_PER_WG)             // 16
#define INF_BITS 0x7F800000u

// Raw IEEE min ops: inputs are provably NaN-free here, so skip the
// sNaN-quieting v_max canonicalizes clang inserts around fminf.
__device__ __forceinline__ float fmin_raw(float a, float b) {
  float r;
  asm("v_min_num_f32 %0, %1, %2" : "=v"(r) : "v"(a), "v"(b));
  return r;
}
__device__ __forceinline__ float fmin3_raw(float a, float b, float c) {
  float r;
  asm("v_min3_num_f32 %0, %1, %2, %3" : "=v"(r) : "v"(a), "v"(b), "v"(c));
  return r;
}

// ---------------------------------------------------------------------------
// Init: both directional min arrays to +inf bits. (2 * 32768 entries)
// ---------------------------------------------------------------------------
__global__ __launch_bounds__(256) void chamfer_init_k(unsigned* __restrict__ bits) {
  int i = blockIdx.x * blockDim.x + threadIdx.x;
  if (i < 2 * B_SZ * N_SZ) bits[i] = INF_BITS;
}

// ---------------------------------------------------------------------------
// Main: d2[m,n] = |x|^2 + |y|^2 - 2 x.y as ONE wmma_f32_16x16x4 per tile pair:
//   A[m][0..2] = x coords, A[m][3] = 1     (K padded)
//   B[0..2][n] = -2 * y coords, B[3][n] = |y|^2
//   C[m][n]    = |x|^2   (per-row -> built ONCE per wave, loop-invariant)
// Grid: (x-groups, batch, y-segment). Both row and column mins flushed via
// clamped-uint-bit atomic min (exact & commutative -> deterministic).
// ---------------------------------------------------------------------------
__global__ __launch_bounds__(256) void chamfer_main_k(const float* __restrict__ X,
                                                      const float* __restrict__ Y,
                                                      unsigned* __restrict__ d1bits,
                                                      unsigned* __restrict__ d2bits) {
  __shared__ float4   sy[CHUNK];       // staged Y segment {x,y,z,|y|^2} (16 KB)
  __shared__ unsigned scol[CHUNK];     // per-segment column mins (4 KB)

  const int  tid  = threadIdx.x;
  const int  lane = tid & 31;
  const int  wave = tid >> 5;
  const bool hi   = lane >= 16;
  const int  b    = blockIdx.y;
  const int  c0   = blockIdx.z * CHUNK;            // Y segment base
  const int  rowBase = blockIdx.x * ROWS_PER_WG + wave * (TPW * 16);

  const float* __restrict__ Xb = X + (size_t)b * N_SZ * 3;
  const float* __restrict__ Yb = Y + (size_t)b * M_SZ * 3;

  // Loop-invariant operands:
  //   A (f32 16x4 layout): lanes 0-15 hold K=(0,1)=(x,y); lanes 16-31 K=(2,3)=(z,1).
  //   C: VGPR v = x2[rowBase+t*16+v] (lanes 0-15) / x2[...+8+v] (lanes 16-31),
  //      gathered from the per-lane x2 via 8 shuffles per tile (one-time cost).
  v2f A[TPW];
  v8f Cx[TPW];
#pragma unroll
  for (int t = 0; t < TPW; ++t) {
    int r = rowBase + t * 16 + (lane & 15);
    float x = Xb[r * 3 + 0];
    float y = Xb[r * 3 + 1];
    float z = Xb[r * 3 + 2];
    float n2 = fmaf(x, x, fmaf(y, y, z * z));
    A[t][0] = hi ? z    : x;
    A[t][1] = hi ? 1.0f : y;
#pragma unroll
    for (int v = 0; v < 8; ++v)
      Cx[t][v] = __shfl(n2, (hi ? 8 : 0) + v, 32);
  }

  const float INF = __builtin_inff();
  v8f rowAcc[TPW];
#pragma unroll
  for (int t = 0; t < TPW; ++t)
#pragma unroll
    for (int v = 0; v < 8; ++v) rowAcc[t][v] = INF;

  // Stage Y segment into LDS as {x,y,z,|y|^2} + init column mins.
  for (int p = tid; p < CHUNK; p += 256) {
    const float* yp = Yb + (size_t)(c0 + p) * 3;
    float yx = yp[0], yy = yp[1], yz = yp[2];
    float4 v;
    v.x = yx; v.y = yy; v.z = yz;
    v.w = fmaf(yx, yx, fmaf(yy, yy, yz * yz));
    sy[p] = v;
  }
  for (int j = tid; j < CHUNK; j += 256) scol[j] = INF_BITS;
  __syncthreads();

  for (int tile = 0; tile < CHUNK / 16; ++tile) {
    // One conflict-free b128 load: lanes 0-15 cover all 64 banks, 16-31 broadcast.
    float4 yv = sy[tile * 16 + (lane & 15)];

    // B operand: lanes 0-15 rows K=(0,1); lanes 16-31 rows K=(2, 3=|y|^2).
    v2f Bop;
    Bop[0] = hi ? (-2.0f * yv.z) : (-2.0f * yv.x);
    Bop[1] = hi ? yv.w           : (-2.0f * yv.y);

    float cm = INF;
#pragma unroll
    for (int t = 0; t < TPW; ++t) {
      v8f d = __builtin_amdgcn_wmma_f32_16x16x4_f32(
          /*neg_a=*/false, A[t], /*neg_b=*/false, Bop,
          /*c_mod=*/(short)0, Cx[t], /*reuse_a=*/false, /*reuse_b=*/false);
#pragma unroll
      for (int v = 0; v < 8; ++v)
        rowAcc[t][v] = fmin_raw(rowAcc[t][v], d[v]);
      cm = fmin3_raw(cm, d[0], d[1]);
      cm = fmin3_raw(cm, d[2], d[3]);
      cm = fmin3_raw(cm, d[4], d[5]);
      cm = fmin3_raw(cm, d[6], d[7]);
    }
    // Combine halves: lanes l and l+16 hold the same column, rows 0-7 / 8-15.
    cm = fmin_raw(cm, __shfl_xor(cm, 16, 32));
    cm = fmaxf(cm, 0.0f);  // clamp so uint-bit ordering == float ordering
    // Both halves hold identical cm -> uniform (non-divergent) idempotent min.
    atomicMin(&scol[tile * 16 + (lane & 15)], __float_as_uint(cm));
  }
  __syncthreads();

  // Flush column mins for this segment to global.
  for (int j = tid; j < CHUNK; j += 256)
    atomicMin(&d2bits[(size_t)b * M_SZ + c0 + j], scol[j]);

  // Row mins: butterfly min within each 16-lane half, then lane-select so
  // every lane carries the min of row (rowBase + t*16 + (lane&7) + 8*hi).
#pragma unroll
  for (int t = 0; t < TPW; ++t) {
    float myval = INF;
#pragma unroll
    for (int v = 0; v < 8; ++v) {
      float val = rowAcc[t][v];
      val = fmin_raw(val, __shfl_xor(val, 8, 32));
      val = fmin_raw(val, __shfl_xor(val, 4, 32));
      val = fmin_raw(val, __shfl_xor(val, 2, 32));
      val = fmin_raw(val, __shfl_xor(val, 1, 32));
      myval = ((lane & 7) == v) ? val : myval;
    }
    int row = rowBase + t * 16 + (lane & 7) + (hi ? 8 : 0);
    myval = fmaxf(myval, 0.0f);
    atomicMin(&d1bits[(size_t)b * N_SZ + row], __float_as_uint(myval));
  }
}

// ---------------------------------------------------------------------------
// Final deterministic reduce: one block, fixed-tree. B*N == B*M == 32768.
// ---------------------------------------------------------------------------
__global__ __launch_bounds__(1024) void chamfer_reduce_k(const unsigned* __restrict__ d1bits,
                                                         const unsigned* __restrict__ d2bits,
                                                         float* __restrict__ out) {
  __shared__ float sm[1024];
  const int tid = threadIdx.x;
  float s = 0.0f;
  for (int j = tid; j < B_SZ * N_SZ; j += 1024)
    s += __uint_as_float(d1bits[j]) + __uint_as_float(d2bits[j]);
  sm[tid] = s;
  __syncthreads();
  for (int w = 512; w > 0; w >>= 1) {
    if (tid < w) sm[tid] += sm[tid + w];
    __syncthreads();
  }
  if (tid == 0) out[0] = sm[0] * (1.0f / (float)(B_SZ * N_SZ));
}

// ---------------------------------------------------------------------------
extern "C" void kernel_launch(void* const* d_in, const int* in_sizes, int n_in,
                              void* d_out, int out_size, void* d_ws, size_t ws_size,
                              hipStream_t stream) {
  (void)in_sizes; (void)n_in; (void)out_size; (void)ws_size;
  const float* X = (const float*)d_in[0];
  const float* Y = (const float*)d_in[1];
  float* out = (float*)d_out;

  unsigned* d1bits = (unsigned*)d_ws;                       // [B*N]
  unsigned* d2bits = d1bits + (size_t)B_SZ * N_SZ;          // [B*M]

  chamfer_init_k<<<(2 * B_SZ * N_SZ + 255) / 256, 256, 0, stream>>>(d1bits);
  dim3 grid(GRID_X, B_SZ, YSEG);          // 16 x 4 x 8 = 512 workgroups
  chamfer_main_k<<<grid, 256, 0, stream>>>(X, Y, d1bits, d2bits);
  chamfer_reduce_k<<<1, 1024, 0, stream>>>(d1bits, d2bits, out);
}